// LlamaAttention_86346022519541
// MI455X (gfx1250) — compile-verified
//
#include <hip/hip_runtime.h>
#include <hip/hip_bf16.h>

typedef float v2f __attribute__((ext_vector_type(2)));
typedef float v4f __attribute__((ext_vector_type(4)));
typedef float v8f __attribute__((ext_vector_type(8)));

#define DIMC  2048
#define NB    8
#define NH    32
#define HDIM  64
#define KCACHE 4095
#define L_TOT 4096

// ---------------------------------------------------------------------------
// out[8][2048] = act[8][2048] @ w[2048][2048]^T  via V_WMMA_F32_16X16X4_F32
// One wave per 16-column tile. A-matrix 16x4 f32 layout (ISA 7.12.2):
//   VGPR0: lanes0-15 K=0, lanes16-31 K=2 ; VGPR1: K=1 / K=3  (M = lane&15)
// B-matrix 4x16 mirrored: lane holds N=lane&15, K pair {2*hi, 2*hi+1}.
// Rows M=8..15 of A are clamped duplicates of rows 0..7; their C rows are
// never stored, so no masking is needed (GEMM rows are independent).
// Weights are single-use -> non-temporal loads.
// ---------------------------------------------------------------------------
__global__ void __launch_bounds__(128)
gemm8_wmma(const float* __restrict__ act, const float* __restrict__ w,
           float* __restrict__ out)
{
  const int lane = threadIdx.x & 31;
  const int wave = threadIdx.x >> 5;
  const int m    = lane & 15;
  const int hi   = lane >> 4;                 // 0 -> K{0,1}, 1 -> K{2,3}
  const int n0   = (blockIdx.x * 4 + wave) * 16;

  const float* ap = act + (size_t)(m & 7) * DIMC + 2 * hi;
  const float* bp = w   + (size_t)(n0 + m) * DIMC + 2 * hi;

  v8f c = {};
  #pragma unroll 8
  for (int k = 0; k < DIMC; k += 4) {
    v2f a = *reinterpret_cast<const v2f*>(ap + k);                       // 8B aligned, reused (cache)
    v2f b = __builtin_nontemporal_load(reinterpret_cast<const v2f*>(bp + k)); // streamed once
    c = __builtin_amdgcn_wmma_f32_16x16x4_f32(false, a, false, b,
                                              (short)0, c, false, false);
  }
  if (hi == 0) {            // lanes 0..15 hold C rows M=0..7 (VGPR i -> M=i)
    #pragma unroll
    for (int i = 0; i < 8; ++i)
      out[(size_t)i * DIMC + n0 + m] = c[i];
  }
}

// ---------------------------------------------------------------------------
// RoPE on q,k + scatter of new-token k,v into the tails of kh / vh.
// qraw/kraw/vraw: [8][2048] as (b, h*64+d).
// kh: [32][8][64][4096], vh: [32][8][4096][64]
// ---------------------------------------------------------------------------
__global__ void __launch_bounds__(256)
rope_scatter(const float* __restrict__ qraw, const float* __restrict__ kraw,
             const float* __restrict__ vraw, const float* __restrict__ fc,
             const float* __restrict__ fs, float* __restrict__ qrope,
             float* __restrict__ kh, float* __restrict__ vh)
{
  const int idx = blockIdx.x * blockDim.x + threadIdx.x;   // 0..16383
  const int b = idx >> 11;
  const int n = idx & 2047;
  const int h = n >> 6;
  const int r = n & 63;
  const int hb = h * NB + b;
  const int j = (r < 32) ? r : (r - 32);
  const float cj = fc[j], sj = fs[j];
  const size_t base = (size_t)b * DIMC + (size_t)h * HDIM;

  const float q0 = qraw[base + 2 * j], q1 = qraw[base + 2 * j + 1];
  const float k0 = kraw[base + 2 * j], k1 = kraw[base + 2 * j + 1];
  const float qv = (r < 32) ? (q0 * cj - q1 * sj) : (q0 * sj + q1 * cj);
  const float kv = (r < 32) ? (k0 * cj - k1 * sj) : (k0 * sj + k1 * cj);

  qrope[base + r] = qv;
  kh[((size_t)hb * HDIM + r) * L_TOT + KCACHE] = kv;
  vh[((size_t)hb * L_TOT + KCACHE) * HDIM + r] = vraw[base + r];
}

// ---------------------------------------------------------------------------
// scores s[hb][l] = dot(q[hb], kh[hb,:,l]) / 8 + mask[b][l]
// Fuses the k_cache -> kh copy into the score read (k_cache read exactly once,
// both sides non-temporal: pure streaming traffic).
// grid = (16 l-tiles, 256 hb), block = 256 (one l per thread).
// ---------------------------------------------------------------------------
__global__ void __launch_bounds__(256)
qk_kernel(const float* __restrict__ qrope, const float* __restrict__ k_cache,
          const float* __restrict__ mask, float* __restrict__ kh,
          float* __restrict__ s_ws)
{
  __shared__ float qs[HDIM];
  const int hb  = blockIdx.y;
  const int b   = hb & 7;
  const int tid = threadIdx.x;
  const int l   = blockIdx.x * 256 + tid;
  if (tid < HDIM) qs[tid] = qrope[(size_t)b * DIMC + (size_t)(hb >> 3) * HDIM + tid];
  __syncthreads();

  const size_t kbase = (size_t)hb * HDIM * KCACHE;
  const size_t obase = (size_t)hb * HDIM * L_TOT;
  float s = 0.f;
  if (l < KCACHE) {
    #pragma unroll 8
    for (int d = 0; d < HDIM; ++d) {
      const float kv = __builtin_nontemporal_load(&k_cache[kbase + (size_t)d * KCACHE + l]);
      __builtin_nontemporal_store(kv, &kh[obase + (size_t)d * L_TOT + l]);  // fused copy
      s = fmaf(qs[d], kv, s);
    }
  } else {                                          // l == 4095: new token
    #pragma unroll 8
    for (int d = 0; d < HDIM; ++d)
      s = fmaf(qs[d], kh[obase + (size_t)d * L_TOT + KCACHE], s);
  }
  s_ws[(size_t)hb * L_TOT + l] = s * 0.125f + mask[(size_t)b * L_TOT + l];
}

// ---------------------------------------------------------------------------
// softmax(s) fused with p @ V and the v_cache -> vh copy. One block per (h,b).
// V path is fully b128: thread owns a 4-wide d-quad, 16 groups stride l.
// ---------------------------------------------------------------------------
__global__ void __launch_bounds__(256)
pv_kernel(const float* __restrict__ s_ws, const float* __restrict__ v_cache,
          float* __restrict__ vh, float* __restrict__ y_ws)
{
  __shared__ float p[L_TOT];        // 16 KB
  __shared__ float red[256];        // 1 KB
  __shared__ float redv[16][HDIM];  // 4 KB (group-partials of y)
  const int hb  = blockIdx.x;
  const int tid = threadIdx.x;
  const size_t sbase = (size_t)hb * L_TOT;

  // --- max ---
  float mx = -3.402823466e38f;
  for (int i = tid; i < L_TOT; i += 256) {
    const float v = s_ws[sbase + i];
    p[i] = v;
    mx = fmaxf(mx, v);
  }
  red[tid] = mx; __syncthreads();
  for (int off = 128; off > 0; off >>= 1) {
    if (tid < off) red[tid] = fmaxf(red[tid], red[tid + off]);
    __syncthreads();
  }
  mx = red[0]; __syncthreads();

  // --- exp & sum ---
  float sum = 0.f;
  for (int i = tid; i < L_TOT; i += 256) {
    const float e = __expf(p[i] - mx);
    p[i] = e;
    sum += e;
  }
  red[tid] = sum; __syncthreads();
  for (int off = 128; off > 0; off >>= 1) {
    if (tid < off) red[tid] += red[tid + off];
    __syncthreads();
  }
  const float inv = 1.f / red[0]; __syncthreads();

  // --- p @ V with fused v_cache -> vh copy (b128 both sides, non-temporal) ---
  const int q = tid & 15;            // d-quad: covers d = 4q .. 4q+3
  const int g = tid >> 4;            // 16 row-groups, l stride 16
  const size_t vbase = (size_t)hb * KCACHE * HDIM + 4 * q;
  const size_t obase = (size_t)hb * L_TOT  * HDIM + 4 * q;
  v4f acc = {};
  for (int l = g; l < KCACHE; l += 16) {
    const v4f vv = __builtin_nontemporal_load(
        reinterpret_cast<const v4f*>(&v_cache[vbase + (size_t)l * HDIM]));
    __builtin_nontemporal_store(vv,
        reinterpret_cast<v4f*>(&vh[obase + (size_t)l * HDIM]));   // fused copy
    acc += p[l] * vv;
  }
  if (g == 15) {                     // 4095 % 16 == 15: new-token row (already in vh)
    const v4f vv = *reinterpret_cast<const v4f*>(&vh[obase + (size_t)KCACHE * HDIM]);
    acc += p[KCACHE] * vv;
  }

  *reinterpret_cast<v4f*>(&redv[g][4 * q]) = acc;
  __syncthreads();
  if (tid < HDIM) {
    float y = 0.f;
    #pragma unroll
    for (int gg = 0; gg < 16; ++gg) y += redv[gg][tid];
    const int h = hb >> 3, b = hb & 7;
    y_ws[(size_t)b * DIMC + (size_t)h * HDIM + tid] = y * inv;
  }
}

// ---------------------------------------------------------------------------
extern "C" void kernel_launch(void* const* d_in, const int* in_sizes, int n_in,
                              void* d_out, int out_size, void* d_ws, size_t ws_size,
                              hipStream_t stream)
{
  (void)in_sizes; (void)n_in; (void)out_size; (void)ws_size;
  const float* hidden  = (const float*)d_in[0];
  const float* fc      = (const float*)d_in[1];
  const float* fs      = (const float*)d_in[2];
  const float* mask    = (const float*)d_in[3];
  const float* k_cache = (const float*)d_in[4];
  const float* v_cache = (const float*)d_in[5];
  const float* wq      = (const float*)d_in[6];
  const float* wk      = (const float*)d_in[7];
  const float* wv      = (const float*)d_in[8];
  const float* wo      = (const float*)d_in[9];

  float* y_out = (float*)d_out;                       // [8][2048]
  float* kh    = y_out + (size_t)NB * DIMC;           // [32][8][64][4096]
  float* vh    = kh + (size_t)NH * NB * HDIM * L_TOT; // [32][8][4096][64]

  float* ws    = (float*)d_ws;
  float* qraw  = ws;
  float* kraw  = qraw  + (size_t)NB * DIMC;
  float* vraw  = kraw  + (size_t)NB * DIMC;
  float* qrope = vraw  + (size_t)NB * DIMC;
  float* y_ws  = qrope + (size_t)NB * DIMC;
  float* s_ws  = y_ws  + (size_t)NB * DIMC;           // [256][4096] scores

  const dim3 blk128(128), blk256(256);
  gemm8_wmma<<<dim3(32), blk128, 0, stream>>>(hidden, wq, qraw);
  gemm8_wmma<<<dim3(32), blk128, 0, stream>>>(hidden, wk, kraw);
  gemm8_wmma<<<dim3(32), blk128, 0, stream>>>(hidden, wv, vraw);
  rope_scatter<<<dim3(64), blk256, 0, stream>>>(qraw, kraw, vraw, fc, fs,
                                                qrope, kh, vh);
  qk_kernel<<<dim3(16, 256), blk256, 0, stream>>>(qrope, k_cache, mask, kh, s_ws);
  pv_kernel<<<dim3(256), blk256, 0, stream>>>(s_ws, v_cache, vh, y_ws);
  gemm8_wmma<<<dim3(32), blk128, 0, stream>>>(y_ws, wo, y_out);
}